// distanceNet_21388937134368
// MI455X (gfx1250) — compile-verified
//
#include <hip/hip_runtime.h>

#define N_NODES 10000
#define N_EDGES 320000
#define D_FEAT  128

// ---------------------------------------------------------------------------
// Pass 0: re-initialize per-node accumulators every call (graph-replay safe).
// seg_max stored as uint bits; ev > 0 always, so 0u == +0.0f is a valid
// identity for max over positive floats.
// ---------------------------------------------------------------------------
__global__ void dn_init_kernel(unsigned int* __restrict__ seg_max,
                               float* __restrict__ seg_sum, int n) {
  int i = blockIdx.x * blockDim.x + threadIdx.x;
  if (i < n) {
    seg_max[i] = 0u;
    seg_sum[i] = 0.0f;
  }
}

// ---------------------------------------------------------------------------
// Pass 1: one wave32 per edge. A 128-float feature row is exactly
// 32 lanes x float4 = 512B -> one perfectly coalesced global_load_b128 per
// row per wave, L2-resident. Edge indices are wave-uniform: force them onto
// the scalar path (s_load / constant cache) via readfirstlane so the vector
// memory pipe only carries the two row gathers. Butterfly reduce
// (ds swizzle), then lane 0 does v_exp_f32 and a native
// global_atomic_max_u32 on the positive-float bit pattern.
// ---------------------------------------------------------------------------
__global__ void dn_edge_l1max_kernel(const float* __restrict__ feats,
                                     const int* __restrict__ edge_index,
                                     float* __restrict__ e_buf,
                                     unsigned int* __restrict__ seg_max) {
  const int lane = threadIdx.x & 31;
  const int wave = threadIdx.x >> 5;
  const int e_v  = blockIdx.x * (blockDim.x >> 5) + wave;
  if (e_v >= N_EDGES) return;

  // wave-uniform edge id -> scalar registers -> scalar loads of src/dst
  const int e   = __builtin_amdgcn_readfirstlane(e_v);
  const int src = __builtin_amdgcn_readfirstlane(edge_index[e]);            // row 0
  const int dst = __builtin_amdgcn_readfirstlane(edge_index[N_EDGES + e]);  // row 1

  const float4* ps = reinterpret_cast<const float4*>(feats + (size_t)src * D_FEAT) + lane;
  const float4* pd = reinterpret_cast<const float4*>(feats + (size_t)dst * D_FEAT) + lane;
  const float4 a = *ps;
  const float4 b = *pd;

  float s = fabsf(a.x - b.x) + fabsf(a.y - b.y) +
            fabsf(a.z - b.z) + fabsf(a.w - b.w);

  // wave32 butterfly reduction over the 32 lanes
  #pragma unroll
  for (int off = 16; off > 0; off >>= 1)
    s += __shfl_xor(s, off, 32);

  if (lane == 0) {
    const float ev = __expf(-0.01f * s);
    e_buf[e] = ev;
    atomicMax(seg_max + dst, __float_as_uint(ev));  // global_atomic_max_u32
  }
}

// ---------------------------------------------------------------------------
// Pass 2: one thread per edge. ex = exp(ev - max[dst]); accumulate per-node
// denominator with native global_atomic_add_f32. e_buf is overwritten in
// place with ex (same slot, deterministic).
// ---------------------------------------------------------------------------
__global__ void dn_edge_expsum_kernel(const int* __restrict__ edge_index,
                                      float* __restrict__ e_buf,
                                      const unsigned int* __restrict__ seg_max,
                                      float* __restrict__ seg_sum) {
  const int e = blockIdx.x * blockDim.x + threadIdx.x;
  if (e >= N_EDGES) return;

  const int dst = edge_index[N_EDGES + e];
  const float m = __uint_as_float(seg_max[dst]);
  const float ex = __expf(e_buf[e] - m);
  e_buf[e] = ex;
  atomicAdd(seg_sum + dst, ex);                     // global_atomic_add_f32
}

// ---------------------------------------------------------------------------
// Pass 3: out[e] = ex[e] / sum[dst]. Pure streaming, L2-resident divisor.
// ---------------------------------------------------------------------------
__global__ void dn_edge_norm_kernel(const int* __restrict__ edge_index,
                                    const float* __restrict__ e_buf,
                                    const float* __restrict__ seg_sum,
                                    float* __restrict__ out) {
  const int e = blockIdx.x * blockDim.x + threadIdx.x;
  if (e >= N_EDGES) return;

  const int dst = edge_index[N_EDGES + e];
  out[e] = e_buf[e] / seg_sum[dst];
}

extern "C" void kernel_launch(void* const* d_in, const int* in_sizes, int n_in,
                              void* d_out, int out_size, void* d_ws, size_t ws_size,
                              hipStream_t stream) {
  const float* feats      = reinterpret_cast<const float*>(d_in[0]);
  const int*   edge_index = reinterpret_cast<const int*>(d_in[1]);
  float*       out        = reinterpret_cast<float*>(d_out);

  // workspace layout (all 4-byte aligned): e_buf[E] | seg_max[N] | seg_sum[N]
  float*        e_buf   = reinterpret_cast<float*>(d_ws);
  unsigned int* seg_max = reinterpret_cast<unsigned int*>(e_buf + N_EDGES);
  float*        seg_sum = reinterpret_cast<float*>(seg_max + N_NODES);

  // Pass 0: init accumulators (10000 nodes)
  {
    const int threads = 256;
    const int blocks  = (N_NODES + threads - 1) / threads;
    dn_init_kernel<<<blocks, threads, 0, stream>>>(seg_max, seg_sum, N_NODES);
  }

  // Pass 1: wave-per-edge L1 distance + exp + segment max
  {
    const int threads = 256;                    // 8 wave32 = 8 edges per block
    const int edges_per_block = threads / 32;
    const int blocks = (N_EDGES + edges_per_block - 1) / edges_per_block;
    dn_edge_l1max_kernel<<<blocks, threads, 0, stream>>>(feats, edge_index,
                                                         e_buf, seg_max);
  }

  // Pass 2: thread-per-edge renormalized exp + segment sum
  {
    const int threads = 256;
    const int blocks  = (N_EDGES + threads - 1) / threads;
    dn_edge_expsum_kernel<<<blocks, threads, 0, stream>>>(edge_index, e_buf,
                                                          seg_max, seg_sum);
  }

  // Pass 3: thread-per-edge final division
  {
    const int threads = 256;
    const int blocks  = (N_EDGES + threads - 1) / threads;
    dn_edge_norm_kernel<<<blocks, threads, 0, stream>>>(edge_index, e_buf,
                                                        seg_sum, out);
  }
}